// ModelCon_62139586839142
// MI455X (gfx1250) — compile-verified
//
#include <hip/hip_runtime.h>
#include <math.h>

// ---------------------------------------------------------------------------
// Model constants (from the reference)
// ---------------------------------------------------------------------------
static constexpr int   BSZ    = 1024;
static constexpr int   CCH    = 8192;
static constexpr int   PROT   = 1218;
static constexpr int   XROW   = PROT + CCH;       // 9410
static constexpr int   XSTRIDE= 2 * XROW;         // 18820 floats per batch row
static constexpr int   ADJN   = 2 * CCH;          // 16384
static constexpr int   SONSEN = 2 * PROT;         // 2436
static constexpr int   MERGED = 2 * PROT + 2 * 64 + 3; // 2567
static constexpr float NEGMASK= -4294967295.0f;   // -2^32 + 1

static constexpr int ACT_NONE = 0;
static constexpr int ACT_RELU = 1;
static constexpr int ACT_LRELU= 2;
static constexpr int ACT_ELU  = 3;

typedef float v2f __attribute__((ext_vector_type(2)));
typedef float v8f __attribute__((ext_vector_type(8)));

// ---------------------------------------------------------------------------
// fp32 WMMA GEMM:  C = act(A @ W + bias)
//   A: M x K row-major (leading dim lda), W: K x N row-major (ld == N).
// REQUIRES: M % 64 == 0 and N % 64 == 0 (true at every call site), so the hot
// loop has NO bounds checks -> unconditional loads + v_wmma only. The K tail
// (K % 4 != 0) is handled once, outside the loop, with guarded loads.
//
// Register blocking: each wave owns a 16x64 output strip (4 accumulators);
// one A fragment feeds 4 WMMAs per k-step (4x A reuse). 128-thread block =
// 4 waves stacked in M -> 64x64 block tile.
//
// Software pipelining: next k-step's fragments (1x b64 A load + 8x b32 B
// loads) are issued BEFORE the current step's 4 WMMAs, so the WMMA burst
// overlaps the loads instead of draining loadcnt to 0 every instruction.
//
// L2MODE: instead of storing C, accumulate sum(((adj - C) * bmat)^2) into
// *l2acc (fuses the reconstruction loss into the decoder GEMM epilogue).
// ---------------------------------------------------------------------------
template <int ACT, bool L2MODE>
__global__ __launch_bounds__(128)
void wmma_gemm(const float* __restrict__ A, long lda,
               const float* __restrict__ W, int N,
               const float* __restrict__ bias,
               float* __restrict__ C, long ldc,
               int M, int K,
               const float* __restrict__ adj, float* __restrict__ l2acc)
{
    const int lane = threadIdx.x & 31;
    const int wave = threadIdx.x >> 5;
    const int half = lane >> 4;              // 0: lanes 0-15, 1: lanes 16-31
    const int l    = lane & 15;
    const int tm   = (blockIdx.y * 4 + wave) * 16;   // wave's 16 output rows
    const int tn   = blockIdx.x * 64;                // wave's 64 output cols

    // A fragment: lane (half,l) supplies row tm+l, K = k0+2*half, k0+2*half+1
    const float* Ap = A + (long)(tm + l) * lda + half * 2;
    // B fragments: lane (half,l) supplies col tn+t*16+l, same two K values
    const float* Wp = W + (long)(half * 2) * N + tn + l;

    v8f acc[4];
#pragma unroll
    for (int t = 0; t < 4; ++t) acc[t] = (v8f){0.f,0.f,0.f,0.f,0.f,0.f,0.f,0.f};

    const int Kmain = K & ~3;                // multiple of 4 (0 or >= 4)
    if (Kmain > 0) {
        // ---- prologue: load k-step 0 fragments -------------------------
        v2f av, bv[4];
        av.x = Ap[0];
        av.y = Ap[1];
        {
            const float* w0 = Wp;
            const float* w1 = Wp + (long)N;
#pragma unroll
            for (int t = 0; t < 4; ++t) { bv[t].x = w0[t * 16]; bv[t].y = w1[t * 16]; }
        }
        // ---- pipelined main loop ---------------------------------------
#pragma unroll 2
        for (int k0 = 0; k0 < Kmain - 4; k0 += 4) {
            Ap += 4;
            Wp += 4 * (long)N;
            v2f an, bn[4];
            an.x = Ap[0];
            an.y = Ap[1];
            const float* w0 = Wp;
            const float* w1 = Wp + (long)N;
#pragma unroll
            for (int t = 0; t < 4; ++t) { bn[t].x = w0[t * 16]; bn[t].y = w1[t * 16]; }
#pragma unroll
            for (int t = 0; t < 4; ++t)
                acc[t] = __builtin_amdgcn_wmma_f32_16x16x4_f32(
                             false, av, false, bv[t], (short)0, acc[t], false, false);
            av = an;
#pragma unroll
            for (int t = 0; t < 4; ++t) bv[t] = bn[t];
        }
        // ---- drain last main k-step ------------------------------------
#pragma unroll
        for (int t = 0; t < 4; ++t)
            acc[t] = __builtin_amdgcn_wmma_f32_16x16x4_f32(
                         false, av, false, bv[t], (short)0, acc[t], false, false);
        Ap += 4;
        Wp += 4 * (long)N;
    }

    if (Kmain < K) {                         // K tail (runs at most once)
        const int  ka  = Kmain + half * 2;
        const bool ok0 = ka < K, ok1 = ka + 1 < K;
        v2f av;
        av.x = ok0 ? Ap[0] : 0.f;
        av.y = ok1 ? Ap[1] : 0.f;
        const float* w0 = Wp;
        const float* w1 = Wp + (long)N;
#pragma unroll
        for (int t = 0; t < 4; ++t) {
            v2f bv;
            bv.x = ok0 ? w0[t * 16] : 0.f;
            bv.y = ok1 ? w1[t * 16] : 0.f;
            acc[t] = __builtin_amdgcn_wmma_f32_16x16x4_f32(
                         false, av, false, bv, (short)0, acc[t], false, false);
        }
    }

    float l2local = 0.f;
#pragma unroll
    for (int t = 0; t < 4; ++t) {
        const int   c  = tn + t * 16 + l;
        const float bz = bias[c];
#pragma unroll
        for (int v = 0; v < 8; ++v) {
            const int r = tm + half * 8 + v;   // C/D layout: VGPR v -> row half*8+v
            float x = acc[t][v] + bz;
            if (ACT == ACT_RELU)  x = fmaxf(x, 0.f);
            if (ACT == ACT_LRELU) x = (x > 0.f) ? x : 0.01f * x;
            if (ACT == ACT_ELU)   x = (x > 0.f) ? x : (expf(x) - 1.f);
            if (L2MODE) {
                const float a  = adj[(long)r * N + c];
                const float bm = (a != 0.f) ? 5.0f : 1.0f;   // BETA
                const float d  = (a - x) * bm;
                l2local += d * d;
            } else {
                C[(long)r * ldc + c] = x;
            }
        }
    }

    if (L2MODE) {
        __shared__ float red[128];
        red[threadIdx.x] = l2local;
        __syncthreads();
        for (int s = 64; s > 0; s >>= 1) {
            if (threadIdx.x < s) red[threadIdx.x] += red[threadIdx.x + s];
            __syncthreads();
        }
        if (threadIdx.x == 0) atomicAdd(l2acc, red[0]);
    }
}

// ---------------------------------------------------------------------------
// Small utility / VALU kernels
// ---------------------------------------------------------------------------
__global__ void zero_kernel(float* p, int n) {
    int i = blockIdx.x * blockDim.x + threadIdx.x;
    if (i < n) p[i] = 0.f;
}

// Build adj (B x 16384) = concat(adj1, adj2) from X
__global__ void copy_adj(const float* __restrict__ X, float* __restrict__ adj) {
    long i = (long)blockIdx.x * blockDim.x + threadIdx.x;
    if (i >= (long)BSZ * ADJN) return;
    int b = (int)(i >> 14);
    int k = (int)(i & (ADJN - 1));
    const float* row = X + (long)b * XSTRIDE;
    adj[i] = (k < CCH) ? row[PROT + k] : row[XROW + PROT + (k - CCH)];
}

// Global sum / sumsq reduction (atomic into out2[0], out2[1])
__global__ __launch_bounds__(256)
void reduce_stats(const float* __restrict__ x, int n, float* out2) {
    float s = 0.f, ss = 0.f;
    for (int i = blockIdx.x * blockDim.x + threadIdx.x; i < n;
         i += gridDim.x * blockDim.x) {
        float v = x[i]; s += v; ss += v * v;
    }
    __shared__ float r1[256], r2[256];
    r1[threadIdx.x] = s; r2[threadIdx.x] = ss;
    __syncthreads();
    for (int st = 128; st > 0; st >>= 1) {
        if (threadIdx.x < st) { r1[threadIdx.x] += r1[threadIdx.x + st];
                                r2[threadIdx.x] += r2[threadIdx.x + st]; }
        __syncthreads();
    }
    if (threadIdx.x == 0) { atomicAdd(out2, r1[0]); atomicAdd(out2 + 1, r2[0]); }
}

// y = conv3x3_SAME(bn(x)) for B images of 1x32x32 (single channel)
__global__ __launch_bounds__(256)
void conv3_bn(const float* __restrict__ x, float* __restrict__ y,
              const float* __restrict__ stats, float n_inv,
              const float* __restrict__ g, const float* __restrict__ b,
              const float* __restrict__ w9)
{
    const int bb = blockIdx.x;
    __shared__ float t[32][33];
    __shared__ float wsh[9];
    const float mean = stats[0] * n_inv;
    const float var  = stats[1] * n_inv - mean * mean;
    const float sc   = rsqrtf(var + 1e-5f) * g[0];
    const float of   = b[0] - mean * sc;
    if (threadIdx.x < 9) wsh[threadIdx.x] = w9[threadIdx.x];
    for (int i = threadIdx.x; i < 1024; i += blockDim.x)
        t[i >> 5][i & 31] = x[(long)bb * 1024 + i] * sc + of;
    __syncthreads();
    for (int i = threadIdx.x; i < 1024; i += blockDim.x) {
        const int r = i >> 5, c = i & 31;
        float acc = 0.f;
#pragma unroll
        for (int di = 0; di < 3; ++di)
#pragma unroll
            for (int dj = 0; dj < 3; ++dj) {
                const int rr = r + di - 1, cc = c + dj - 1;
                if (rr >= 0 && rr < 32 && cc >= 0 && cc < 32)
                    acc += wsh[di * 3 + dj] * t[rr][cc];
            }
        y[(long)bb * 1024 + i] = acc;
    }
}

// y = relu(bn(x)) elementwise
__global__ void bn_relu(const float* __restrict__ x, float* __restrict__ y,
                        const float* __restrict__ stats, float n_inv,
                        const float* __restrict__ g, const float* __restrict__ b,
                        int n)
{
    int i = blockIdx.x * blockDim.x + threadIdx.x;
    if (i >= n) return;
    const float mean = stats[0] * n_inv;
    const float var  = stats[1] * n_inv - mean * mean;
    const float sc   = rsqrtf(var + 1e-5f) * g[0];
    const float of   = b[0] - mean * sc;
    y[i] = fmaxf(x[i] * sc + of, 0.f);
}

// Per-column mean / mean-of-squares over M rows. grid.x == N columns.
__global__ __launch_bounds__(256)
void col_stats(const float* __restrict__ x, int M, int ld, float* __restrict__ out)
{
    const int j = blockIdx.x;
    __shared__ float r1[256], r2[256];
    float s = 0.f, ss = 0.f;
    for (int i = threadIdx.x; i < M; i += 256) {
        float v = x[(long)i * ld + j]; s += v; ss += v * v;
    }
    r1[threadIdx.x] = s; r2[threadIdx.x] = ss;
    __syncthreads();
    for (int st = 128; st > 0; st >>= 1) {
        if (threadIdx.x < st) { r1[threadIdx.x] += r1[threadIdx.x + st];
                                r2[threadIdx.x] += r2[threadIdx.x + st]; }
        __syncthreads();
    }
    if (threadIdx.x == 0) {
        out[2 * j]     = r1[0] / (float)M;
        out[2 * j + 1] = r2[0] / (float)M;
    }
}

// _scale: x = (x - mean) / (std == 0 ? 1 : std), std biased, per column
__global__ void scale_cols(float* __restrict__ x, int M, int N, int ld,
                           const float* __restrict__ st)
{
    long i = (long)blockIdx.x * blockDim.x + threadIdx.x;
    if (i >= (long)M * N) return;
    const int r = (int)(i / N), c = (int)(i % N);
    const float m  = st[2 * c];
    const float sd = sqrtf(fmaxf(st[2 * c + 1] - m * m, 0.f));
    const float d  = (sd == 0.f) ? 1.f : sd;
    x[(long)r * ld + c] = (x[(long)r * ld + c] - m) / d;
}

// bn1d: x = (x - mean) * rsqrt(var + 1e-5) * g[c] + b[c], in place
__global__ void bn_cols(float* __restrict__ x, int M, int N,
                        const float* __restrict__ st,
                        const float* __restrict__ g, const float* __restrict__ b)
{
    long i = (long)blockIdx.x * blockDim.x + threadIdx.x;
    if (i >= (long)M * N) return;
    const int r = (int)(i / N), c = (int)(i % N);
    const float m = st[2 * c];
    const float v = st[2 * c + 1] - m * m;
    x[(long)r * N + c] = (x[(long)r * N + c] - m) * rsqrtf(v + 1e-5f) * g[c] + b[c];
}

// L2-normalize each row of concat(p1,p2) gathered from X, write into the
// pre-merged buffer at column offset 128.
__global__ __launch_bounds__(256)
void sonsen_l2(const float* __restrict__ X, float* __restrict__ pm)
{
    const int bb = blockIdx.x;
    const float* x1 = X + (long)bb * XSTRIDE;
    const float* x2 = x1 + XROW;
    __shared__ float red[256];
    float s = 0.f;
    for (int c = threadIdx.x; c < SONSEN; c += 256) {
        const float v = (c < PROT) ? x1[c] : x2[c - PROT];
        s += v * v;
    }
    red[threadIdx.x] = s;
    __syncthreads();
    for (int st = 128; st > 0; st >>= 1) {
        if (threadIdx.x < st) red[threadIdx.x] += red[threadIdx.x + st];
        __syncthreads();
    }
    const float n   = sqrtf(red[0]);
    const float inv = (n == 0.f) ? 1.f : 1.f / n;
    for (int c = threadIdx.x; c < SONSEN; c += 256) {
        const float v = (c < PROT) ? x1[c] : x2[c - PROT];
        pm[(long)bb * MERGED + 128 + c] = v * inv;
    }
}

// In-place row L2 normalization (rows of length N, leading dim ld)
__global__ __launch_bounds__(256)
void l2rows(float* __restrict__ x, int N, int ld)
{
    float* row = x + (long)blockIdx.x * ld;
    __shared__ float red[256];
    float s = 0.f;
    for (int c = threadIdx.x; c < N; c += 256) { float v = row[c]; s += v * v; }
    red[threadIdx.x] = s;
    __syncthreads();
    for (int st = 128; st > 0; st >>= 1) {
        if (threadIdx.x < st) red[threadIdx.x] += red[threadIdx.x + st];
        __syncthreads();
    }
    const float n   = sqrtf(red[0]);
    const float inv = (n == 0.f) ? 1.f : 1.f / n;
    for (int c = threadIdx.x; c < N; c += 256) row[c] *= inv;
}

// Full tiny attention: q/k/v in (B,32,32); H=4 heads of dim 8; mask, softmax,
// residual, layernorm (ddof=1). One block per batch element.
__global__ __launch_bounds__(256)
void attn_kernel(const float* __restrict__ qin, const float* __restrict__ kin,
                 const float* __restrict__ vin,
                 const float* __restrict__ qw, const float* __restrict__ qb,
                 const float* __restrict__ kw, const float* __restrict__ kb,
                 const float* __restrict__ vw, const float* __restrict__ vb,
                 const float* __restrict__ lng, const float* __restrict__ lnb,
                 float* __restrict__ out)
{
    const int bb = blockIdx.x;
    const float* Qi = qin + (long)bb * 1024;
    const float* Ki = kin + (long)bb * 1024;
    const float* Vi = vin + (long)bb * 1024;
    __shared__ float Q[32][33], K[32][33], V[32][33], O[32][33];
    __shared__ float qm[32], km[32];
    const int t = threadIdx.x;

    if (t < 32) {
        float sq = 0.f, sk = 0.f;
        for (int c = 0; c < 32; ++c) { sq += Qi[t * 32 + c]; sk += Ki[t * 32 + c]; }
        qm[t] = (sq != 0.f) ? 1.f : 0.f;
        km[t] = (sk != 0.f) ? 1.f : 0.f;
    }
    for (int idx = t; idx < 1024; idx += 256) {
        const int i = idx >> 5, j = idx & 31;
        float aq = qb[j], ak = kb[j], av = vb[j];
        for (int c = 0; c < 32; ++c) {
            aq += Qi[i * 32 + c] * qw[c * 32 + j];
            ak += Ki[i * 32 + c] * kw[c * 32 + j];
            av += Vi[i * 32 + c] * vw[c * 32 + j];
        }
        Q[i][j] = fmaxf(aq, 0.f);
        K[i][j] = fmaxf(ak, 0.f);
        V[i][j] = fmaxf(av, 0.f);
    }
    __syncthreads();
    if (t < 128) {                        // 4 heads x 32 query rows
        const int h = t >> 5, i = t & 31;
        float sc[32];
        float mx = -3.402823466e38f;
        for (int j = 0; j < 32; ++j) {
            float s = 0.f;
            for (int d = 0; d < 8; ++d) s += Q[i][h * 8 + d] * K[j][h * 8 + d];
            s *= 0.35355339059327373f;    // 1/sqrt(8)
            if (km[j] == 0.f) s = NEGMASK;
            sc[j] = s;
            mx = fmaxf(mx, s);
        }
        float den = 0.f;
        for (int j = 0; j < 32; ++j) { sc[j] = expf(sc[j] - mx); den += sc[j]; }
        const float inv = qm[i] / den;    // softmax * query-mask, folded
        for (int d = 0; d < 8; ++d) {
            float o = 0.f;
            for (int j = 0; j < 32; ++j) o += sc[j] * V[j][h * 8 + d];
            O[i][h * 8 + d] = o * inv;
        }
    }
    __syncthreads();
    if (t < 32) {                         // residual + layernorm (ddof=1)
        float row[32]; float m = 0.f;
        for (int j = 0; j < 32; ++j) { row[j] = O[t][j] + Qi[t * 32 + j]; m += row[j]; }
        m *= (1.f / 32.f);
        float vv = 0.f;
        for (int j = 0; j < 32; ++j) { const float d = row[j] - m; vv += d * d; }
        const float sd  = sqrtf(vv / 31.f);
        const float inv = 1.f / (sd + 1e-8f);
        for (int j = 0; j < 32; ++j)
            out[(long)bb * 1024 + t * 32 + j] = lng[j] * (row[j] - m) * inv + lnb[j];
    }
}

// r = concat(mean_s(pg*a), max_s(pg*a))  ->  (B, 64). grid B, 32 threads.
__global__ __launch_bounds__(32)
void mul_agg(const float* __restrict__ pg, const float* __restrict__ a,
             float* __restrict__ r)
{
    const int bb = blockIdx.x, f = threadIdx.x;
    float s = 0.f, mx = -3.402823466e38f;
    for (int sI = 0; sI < 32; ++sI) {
        const float v = pg[(long)bb * 1024 + sI * 32 + f] *
                        a [(long)bb * 1024 + sI * 32 + f];
        s += v;
        mx = fmaxf(mx, v);
    }
    r[(long)bb * 64 + f]      = s * (1.f / 32.f);
    r[(long)bb * 64 + 32 + f] = mx;
}

// cos / bilinear / tanh-linear pair features + stage r1,r2 into pre-merged
__global__ __launch_bounds__(64)
void pair_feat(const float* __restrict__ r1, const float* __restrict__ r2,
               const float* __restrict__ bilw, const float* __restrict__ lsw,
               const float* __restrict__ lsb, float* __restrict__ pm)
{
    const int bb = blockIdx.x, t = threadIdx.x;
    __shared__ float a[64], b2[64], red[64];
    a[t]  = r1[(long)bb * 64 + t];
    b2[t] = r2[(long)bb * 64 + t];
    pm[(long)bb * MERGED + t]      = a[t];
    pm[(long)bb * MERGED + 64 + t] = b2[t];
    __syncthreads();
    auto reduce = [&](float v) -> float {
        red[t] = v; __syncthreads();
        for (int s = 32; s > 0; s >>= 1) {
            if (t < s) red[t] += red[t + s];
            __syncthreads();
        }
        const float r = red[0]; __syncthreads();
        return r;
    };
    const float dot = reduce(a[t] * b2[t]);
    const float n1  = reduce(a[t] * a[t]);
    const float n2  = reduce(b2[t] * b2[t]);
    float bl = 0.f;
    for (int j = 0; j < 64; ++j) bl += bilw[t * 64 + j] * b2[j];
    const float bil = reduce(a[t] * bl);
    const float lin = reduce(a[t] * lsw[t] + b2[t] * lsw[64 + t]);
    if (t == 0) {
        const float nn1 = fmaxf(sqrtf(n1), 1e-8f);
        const float nn2 = fmaxf(sqrtf(n2), 1e-8f);
        pm[(long)bb * MERGED + 2564] = dot / (nn1 * nn2);
        pm[(long)bb * MERGED + 2565] = bil;
        pm[(long)bb * MERGED + 2566] = tanhf(lin + lsb[0]);
    }
}

// L1 = 2B * sum_i ||emb_i||^2 - 2 * ||sum_i emb_i||^2, from emb column stats
__global__ __launch_bounds__(128)
void compute_L1(const float* __restrict__ st, float* __restrict__ out)
{
    const int t = threadIdx.x;                // 128 embedding columns
    __shared__ float red[128];
    const float Bf = (float)BSZ;
    const float m  = st[2 * t];
    const float ms = st[2 * t + 1];
    red[t] = 2.f * Bf * Bf * ms - 2.f * (Bf * m) * (Bf * m);
    __syncthreads();
    for (int s = 64; s > 0; s >>= 1) {
        if (t < s) red[t] += red[t + s];
        __syncthreads();
    }
    if (t == 0) out[0] = red[0];
}

// h4 = h3 @ cw4 + cb4 for the tiny 128->2 head. grid B, 64 threads.
__global__ __launch_bounds__(64)
void gemm_cw4(const float* __restrict__ h3, const float* __restrict__ w,
              const float* __restrict__ bias, float* __restrict__ h4)
{
    const int bb = blockIdx.x, t = threadIdx.x;
    __shared__ float red[64];
    for (int j = 0; j < 2; ++j) {
        float s = 0.f;
        for (int c = t; c < 128; c += 64) s += h3[(long)bb * 128 + c] * w[c * 2 + j];
        red[t] = s; __syncthreads();
        for (int st2 = 32; st2 > 0; st2 >>= 1) {
            if (t < st2) red[t] += red[t + st2];
            __syncthreads();
        }
        if (t == 0) h4[(long)bb * 2 + j] = red[0] + bias[j];
        __syncthreads();
    }
}

// Final bn1d + sigmoid + argmax
__global__ void final_out(const float* __restrict__ h4, const float* __restrict__ st,
                          const float* __restrict__ g, const float* __restrict__ b,
                          float* __restrict__ out)
{
    const int bb = blockIdx.x * blockDim.x + threadIdx.x;
    if (bb >= BSZ) return;
    float p[2];
#pragma unroll
    for (int j = 0; j < 2; ++j) {
        const float m = st[2 * j];
        const float v = st[2 * j + 1] - m * m;
        const float x = (h4[(long)bb * 2 + j] - m) * rsqrtf(v + 1e-5f) * g[j] + b[j];
        p[j] = 1.f / (1.f + expf(-x));
        out[(long)bb * 2 + j] = p[j];
    }
    out[2 * BSZ + bb] = (p[1] > p[0]) ? 1.f : 0.f;   // argmax (first max wins ties)
}

__global__ void write_losses(const float* __restrict__ L1p,
                             const float* __restrict__ L2p, float* __restrict__ out)
{
    if (threadIdx.x == 0 && blockIdx.x == 0) {
        const float l1 = L1p[0];
        const float l2 = 0.1f * L2p[0];   // ALPHA * L2
        out[3 * BSZ + 0] = l1;
        out[3 * BSZ + 1] = l2;
        out[3 * BSZ + 2] = l1 + l2;
    }
}

// ---------------------------------------------------------------------------
// Host-side GEMM dispatch.  Requires M % 64 == 0, N % 64 == 0.
// ---------------------------------------------------------------------------
static void launch_gemm(hipStream_t s, const float* A, long lda, const float* W,
                        int N, const float* bias, float* C, long ldc,
                        int M, int K, int act)
{
    dim3 g((unsigned)(N / 64), (unsigned)(M / 64)), b(128, 1, 1);
    switch (act) {
        case ACT_RELU:  wmma_gemm<ACT_RELU , false><<<g, b, 0, s>>>(A, lda, W, N, bias, C, ldc, M, K, nullptr, nullptr); break;
        case ACT_LRELU: wmma_gemm<ACT_LRELU, false><<<g, b, 0, s>>>(A, lda, W, N, bias, C, ldc, M, K, nullptr, nullptr); break;
        case ACT_ELU:   wmma_gemm<ACT_ELU  , false><<<g, b, 0, s>>>(A, lda, W, N, bias, C, ldc, M, K, nullptr, nullptr); break;
        default:        wmma_gemm<ACT_NONE , false><<<g, b, 0, s>>>(A, lda, W, N, bias, C, ldc, M, K, nullptr, nullptr); break;
    }
}

// ---------------------------------------------------------------------------
// kernel_launch
// ---------------------------------------------------------------------------
extern "C" void kernel_launch(void* const* d_in, const int* in_sizes, int n_in,
                              void* d_out_v, int out_size, void* d_ws, size_t ws_size,
                              hipStream_t stream)
{
    (void)in_sizes; (void)n_in; (void)out_size; (void)ws_size;

    const float* X    = (const float*)d_in[0];
    const float* glw  = (const float*)d_in[1];
    const float* glb  = (const float*)d_in[2];
    const float* bnag = (const float*)d_in[3];
    const float* bnab = (const float*)d_in[4];
    const float* bn1g = (const float*)d_in[5];
    const float* bn1b = (const float*)d_in[6];
    const float* bnbg = (const float*)d_in[7];
    const float* bnbb = (const float*)d_in[8];
    const float* c1w  = (const float*)d_in[9];
    const float* c2w  = (const float*)d_in[10];
    const float* qw   = (const float*)d_in[11];
    const float* qb   = (const float*)d_in[12];
    const float* kw   = (const float*)d_in[13];
    const float* kb   = (const float*)d_in[14];
    const float* vw   = (const float*)d_in[15];
    const float* vb   = (const float*)d_in[16];
    const float* lng  = (const float*)d_in[17];
    const float* lnb  = (const float*)d_in[18];
    const float* bilw = (const float*)d_in[19];
    const float* lsw  = (const float*)d_in[20];
    const float* lsb  = (const float*)d_in[21];
    const float* e0w  = (const float*)d_in[22];
    const float* e0b  = (const float*)d_in[23];
    const float* e1w  = (const float*)d_in[24];
    const float* e1b  = (const float*)d_in[25];
    const float* d0w  = (const float*)d_in[26];
    const float* d0b  = (const float*)d_in[27];
    const float* d1w  = (const float*)d_in[28];
    const float* d1b  = (const float*)d_in[29];
    const float* cw1  = (const float*)d_in[30];
    const float* cb1  = (const float*)d_in[31];
    const float* cg1  = (const float*)d_in[32];
    const float* cbe1 = (const float*)d_in[33];
    const float* cw2  = (const float*)d_in[34];
    const float* cb2  = (const float*)d_in[35];
    const float* cw3  = (const float*)d_in[36];
    const float* cb3  = (const float*)d_in[37];
    const float* cw4  = (const float*)d_in[38];
    const float* cb4  = (const float*)d_in[39];
    const float* cg2  = (const float*)d_in[40];
    const float* cbe2 = (const float*)d_in[41];
    float* out = (float*)d_out_v;

    // ---- workspace layout (floats) ------------------------------------
    float* wsf = (float*)d_ws;
    size_t off = 0;
    auto take = [&](size_t n) { float* p = wsf + off; off += n; return p; };
    float* adj  = take((size_t)BSZ * ADJN);     // 16.8M  concat adjacency
    float* gl1  = take((size_t)BSZ * 1024);     // p1 @ glw + glb
    float* gl2  = take((size_t)BSZ * 1024);
    float* tA   = take((size_t)BSZ * 1024);     // conv pipeline scratch
    float* tB   = take((size_t)BSZ * 1024);
    float* pg1  = take((size_t)BSZ * 1024);
    float* pg2  = take((size_t)BSZ * 1024);
    float* t0   = take((size_t)BSZ * 1024);     // encoder hidden
    float* emb  = take((size_t)BSZ * 128);
    float* td   = take((size_t)BSZ * 1024);     // decoder hidden
    float* a1   = take((size_t)BSZ * 1024);
    float* a2   = take((size_t)BSZ * 1024);
    float* r1   = take((size_t)BSZ * 64);
    float* r2   = take((size_t)BSZ * 64);
    float* pm   = take((size_t)BSZ * MERGED);   // pre-merged feature matrix
    float* h1   = take((size_t)BSZ * 512);
    float* h2   = take((size_t)BSZ * 256);
    float* h3   = take((size_t)BSZ * 128);
    float* h4   = take((size_t)BSZ * 2);
    float* csb  = take(2 * (size_t)MERGED);     // column-stats scratch (max N)
    float* ces  = take(256);                    // emb column stats
    float* cs2  = take(4);                      // final bn stats (2 cols)
    float* scal = take(16);                     // [0..11] bn2d sums, [12] L2, [13] L1

    const float inv1M = 1.0f / (float)(BSZ * 1024);

    // ---- 0. zero the atomic accumulators (fresh every call) -----------
    zero_kernel<<<1, 32, 0, stream>>>(scal, 16);

    // ---- 1. adjacency copy --------------------------------------------
    copy_adj<<<(BSZ * ADJN) / 256, 256, 0, stream>>>(X, adj);

    // ---- 2. p @ glw + glb (fp32 WMMA, strided reads straight from X) ---
    launch_gemm(stream, X,        XSTRIDE, glw, 1024, glb, gl1, 1024, BSZ, PROT, ACT_NONE);
    launch_gemm(stream, X + XROW, XSTRIDE, glw, 1024, glb, gl2, 1024, BSZ, PROT, ACT_NONE);

    // ---- 3. gated CNN (bn -> conv -> bn-relu -> bn -> conv) ------------
    reduce_stats<<<512, 256, 0, stream>>>(gl1, BSZ * 1024, scal + 0);
    conv3_bn<<<BSZ, 256, 0, stream>>>(gl1, tA, scal + 0, inv1M, bnag, bnab, c1w);
    reduce_stats<<<512, 256, 0, stream>>>(tA, BSZ * 1024, scal + 2);
    bn_relu<<<(BSZ * 1024) / 256, 256, 0, stream>>>(tA, tB, scal + 2, inv1M, bn1g, bn1b, BSZ * 1024);
    reduce_stats<<<512, 256, 0, stream>>>(tB, BSZ * 1024, scal + 4);
    conv3_bn<<<BSZ, 256, 0, stream>>>(tB, pg1, scal + 4, inv1M, bnbg, bnbb, c2w);

    reduce_stats<<<512, 256, 0, stream>>>(gl2, BSZ * 1024, scal + 6);
    conv3_bn<<<BSZ, 256, 0, stream>>>(gl2, tA, scal + 6, inv1M, bnag, bnab, c1w);
    reduce_stats<<<512, 256, 0, stream>>>(tA, BSZ * 1024, scal + 8);
    bn_relu<<<(BSZ * 1024) / 256, 256, 0, stream>>>(tA, tB, scal + 8, inv1M, bn1g, bn1b, BSZ * 1024);
    reduce_stats<<<512, 256, 0, stream>>>(tB, BSZ * 1024, scal + 10);
    conv3_bn<<<BSZ, 256, 0, stream>>>(tB, pg2, scal + 10, inv1M, bnbg, bnbb, c2w);

    // ---- 4. autoencoder (the big WMMA GEMMs) ---------------------------
    launch_gemm(stream, adj, ADJN, e0w, 1024, e0b, t0,  1024, BSZ, ADJN, ACT_LRELU);
    launch_gemm(stream, t0, 1024,  e1w, 128,  e1b, emb, 128,  BSZ, 1024, ACT_LRELU);
    col_stats<<<128, 256, 0, stream>>>(emb, BSZ, 128, ces);
    compute_L1<<<1, 128, 0, stream>>>(ces, scal + 13);
    launch_gemm(stream, emb, 128,  d0w, 1024, d0b, td,  1024, BSZ, 128,  ACT_LRELU);
    // decoder GEMM with fused reconstruction-loss epilogue (no recon store)
    {
        dim3 g(ADJN / 64, BSZ / 64), b(128, 1, 1);
        wmma_gemm<ACT_LRELU, true><<<g, b, 0, stream>>>(
            td, 1024, d1w, ADJN, d1b, nullptr, 0, BSZ, 1024, adj, scal + 12);
    }

    // ---- 5. attention + aggregation ------------------------------------
    attn_kernel<<<BSZ, 256, 0, stream>>>(pg1, pg2, pg1, qw, qb, kw, kb, vw, vb, lng, lnb, a1);
    attn_kernel<<<BSZ, 256, 0, stream>>>(pg2, pg1, pg2, qw, qb, kw, kb, vw, vb, lng, lnb, a2);
    mul_agg<<<BSZ, 32, 0, stream>>>(pg1, a1, r1);
    mul_agg<<<BSZ, 32, 0, stream>>>(pg2, a2, r2);

    // ---- 6. sonsen = scale(l2rows(concat(p1,p2))) into pm[:,128:2564] --
    sonsen_l2<<<BSZ, 256, 0, stream>>>(X, pm);
    col_stats<<<SONSEN, 256, 0, stream>>>(pm + 128, BSZ, MERGED, csb);
    scale_cols<<<((size_t)BSZ * SONSEN + 255) / 256, 256, 0, stream>>>(pm + 128, BSZ, SONSEN, MERGED, csb);

    // ---- 7. pair features + merged = scale(l2rows(pm)) -----------------
    pair_feat<<<BSZ, 64, 0, stream>>>(r1, r2, bilw, lsw, lsb, pm);
    l2rows<<<BSZ, 256, 0, stream>>>(pm, MERGED, MERGED);
    col_stats<<<MERGED, 256, 0, stream>>>(pm, BSZ, MERGED, csb);
    scale_cols<<<((size_t)BSZ * MERGED + 255) / 256, 256, 0, stream>>>(pm, BSZ, MERGED, MERGED, csb);

    // ---- 8. classifier --------------------------------------------------
    launch_gemm(stream, pm, MERGED, cw1, 512, cb1, h1, 512, BSZ, MERGED, ACT_NONE);
    col_stats<<<512, 256, 0, stream>>>(h1, BSZ, 512, csb);
    bn_cols<<<((size_t)BSZ * 512 + 255) / 256, 256, 0, stream>>>(h1, BSZ, 512, csb, cg1, cbe1);
    launch_gemm(stream, h1, 512, cw2, 256, cb2, h2, 256, BSZ, 512, ACT_ELU);
    launch_gemm(stream, h2, 256, cw3, 128, cb3, h3, 128, BSZ, 256, ACT_NONE);
    gemm_cw4<<<BSZ, 64, 0, stream>>>(h3, cw4, cb4, h4);
    col_stats<<<2, 256, 0, stream>>>(h4, BSZ, 2, cs2);
    final_out<<<(BSZ + 255) / 256, 256, 0, stream>>>(h4, cs2, cg2, cbe2, out);

    // ---- 9. losses ------------------------------------------------------
    write_losses<<<1, 32, 0, stream>>>(scal + 13, scal + 12, out);
}